// TabliterModel_23029614641657
// MI455X (gfx1250) — compile-verified
//
#include <hip/hip_runtime.h>
#include <hip/hip_bf16.h>

// out[i] = sigmoid(x[i]*w + b)  — pure HBM-streaming kernel.
// 8 elements/thread via two 128-bit NT loads + two 128-bit NT stores.

typedef float v4f __attribute__((ext_vector_type(4)));

__device__ __forceinline__ float fast_sigmoid(float t) {
    // sigmoid(t) = 1 / (1 + exp(-t)) = 1 / (1 + 2^(-t*log2(e)))
    // v_exp_f32 computes 2^x; v_rcp_f32 finishes it.
    const float LOG2E = 1.4426950408889634f;
    float e = __builtin_amdgcn_exp2f(-t * LOG2E);
    return __builtin_amdgcn_rcpf(1.0f + e);
}

__device__ __forceinline__ v4f sig4(v4f v, float wv, float bv) {
    v4f r;
    r.x = fast_sigmoid(fmaf(v.x, wv, bv));
    r.y = fast_sigmoid(fmaf(v.y, wv, bv));
    r.z = fast_sigmoid(fmaf(v.z, wv, bv));
    r.w = fast_sigmoid(fmaf(v.w, wv, bv));
    return r;
}

__global__ __launch_bounds__(256) void sigmoid_affine_stream(
    const float* __restrict__ x,
    const float* __restrict__ w,
    const float* __restrict__ b,
    float* __restrict__ out,
    long long n)
{
    // Wave-uniform scalars -> s_load_b32, broadcast to all lanes.
    const float wv = w[0];
    const float bv = b[0];

    const long long base =
        ((long long)blockIdx.x * blockDim.x + threadIdx.x) * 8LL;

    if (base + 8 <= n) {
        const v4f* xv = reinterpret_cast<const v4f*>(x + base);
        v4f*       ov = reinterpret_cast<v4f*>(out + base);

        // Two independent 128-bit NT loads -> both in flight before the wait.
        v4f a0 = __builtin_nontemporal_load(xv);
        v4f a1 = __builtin_nontemporal_load(xv + 1);

        v4f r0 = sig4(a0, wv, bv);
        v4f r1 = sig4(a1, wv, bv);

        __builtin_nontemporal_store(r0, ov);
        __builtin_nontemporal_store(r1, ov + 1);
    } else {
        // Scalar tail (not taken for N = 2^25, kept for safety).
        for (long long i = base; i < n; ++i) {
            out[i] = fast_sigmoid(fmaf(x[i], wv, bv));
        }
    }
}

extern "C" void kernel_launch(void* const* d_in, const int* in_sizes, int n_in,
                              void* d_out, int out_size, void* d_ws, size_t ws_size,
                              hipStream_t stream) {
    const float* x = (const float*)d_in[0];  // row_avg, N elements
    const float* w = (const float*)d_in[1];  // 1 element
    const float* b = (const float*)d_in[2];  // 1 element
    float* out = (float*)d_out;

    const long long n = (long long)in_sizes[0];

    const int  threads        = 256;
    const long long perBlock  = (long long)threads * 8;
    const int  blocks         = (int)((n + perBlock - 1) / perBlock);

    sigmoid_affine_stream<<<blocks, threads, 0, stream>>>(x, w, b, out, n);
}